// SimpleCPG_88587995447926
// MI455X (gfx1250) — compile-verified
//
#include <hip/hip_runtime.h>
#include <stdint.h>

// ---------------- constants from the reference ----------------
#define MUF    1.0f
#define DTF    0.01f
#define EPSF   1e-9f
#define BOUNDF 1.5707963267948966f   // pi/2

#define BLK   256                    // 8 waves (wave32)
#define EPT   4                      // edges per thread per chunk
#define CHUNK (BLK * EPT)            // 1024 edges per block-iteration

// ---------------------------------------------------------------
// CDNA5 async global->LDS copy: 16B per lane, tracked by ASYNCcnt.
// GVS mode: mem_addr = SGPR64 base + VGPR32 byte offset.
// LDS address taken from a VGPR (low 32 bits of a flat shared pointer
// are exactly the workgroup-relative LDS byte address per ISA 10.2).
// ---------------------------------------------------------------
__device__ __forceinline__ void async_stage16(uint32_t lds_addr, uint32_t voff,
                                              uint64_t sbase) {
    asm volatile("global_load_async_to_lds_b128 %0, %1, %2"
                 :: "v"(lds_addr), "v"(voff), "s"(sbase) : "memory");
}
__device__ __forceinline__ void wait_async_le2() {
    asm volatile("s_wait_asynccnt 0x2" ::: "memory");
}
__device__ __forceinline__ void wait_async_0() {
    asm volatile("s_wait_asynccnt 0x0" ::: "memory");
}
__device__ __forceinline__ uint32_t lds_off(const void* p) {
    return (uint32_t)(uintptr_t)p;   // truncate flat shared addr -> LDS offset
}

// ---------------------------------------------------------------
// K1: per-node prep. theta = atan2(y, x+eps); we only ever need
// sin/cos(theta) = y/r, (x+eps)/r  -> one rsqrt, no transcendentals.
// Pack {x, y, s, c} into one 16B record (single-line gather later).
// Zero the edge accumulators (must happen every launch).
// ---------------------------------------------------------------
__global__ void node_prep_kernel(const float* __restrict__ x,
                                 const float* __restrict__ y,
                                 float4* __restrict__ pk,
                                 float2* __restrict__ cxy, int n) {
    int i = blockIdx.x * blockDim.x + threadIdx.x;
    if (i >= n) return;
    float xi = x[i], yi = y[i];
    float xe = xi + EPSF;
    float rinv = __frsqrt_rn(xe * xe + yi * yi);
    pk[i]  = make_float4(xi, yi, yi * rinv, xe * rinv);  // {x, y, sin, cos}
    cxy[i] = make_float2(0.f, 0.f);
}

// ---------------------------------------------------------------
// K2: edge pass. Each block-iteration handles CHUNK=1024 edges.
// Each thread async-copies 16B of src-idx and 16B of dst-idx into
// its private LDS slots (double buffered, no cross-thread sharing
// -> no workgroup barriers, only s_wait_asynccnt), then does 4
// edges: two b128 node gathers + one b32 + 2 f32 atomics.
// ---------------------------------------------------------------
__global__ void __launch_bounds__(BLK)
edge_pass_kernel(const int* __restrict__ esrc, const int* __restrict__ edst,
                 const float4* __restrict__ pk, const float* __restrict__ ha,
                 float* __restrict__ cxy, long long nchunks) {
    __shared__ int4 stg_s[2][BLK];
    __shared__ int4 stg_d[2][BLK];

    const int tid = threadIdx.x;
    const uint64_t sbase = (uint64_t)(uintptr_t)esrc;
    const uint64_t dbase = (uint64_t)(uintptr_t)edst;
    const uint32_t ls[2] = { lds_off(&stg_s[0][tid]), lds_off(&stg_s[1][tid]) };
    const uint32_t ld[2] = { lds_off(&stg_d[0][tid]), lds_off(&stg_d[1][tid]) };

    long long chunk  = blockIdx.x;
    const long long cstep = gridDim.x;
    int buf = 0;

    if (chunk < nchunks) {
        uint32_t voff = (uint32_t)((chunk * CHUNK + (long long)tid * EPT) * 4);
        async_stage16(ls[0], voff, sbase);
        async_stage16(ld[0], voff, dbase);
    }

    while (chunk < nchunks) {
        long long cnext = chunk + cstep;
        if (cnext < nchunks) {               // prefetch next chunk into other buffer
            uint32_t voff = (uint32_t)((cnext * CHUNK + (long long)tid * EPT) * 4);
            async_stage16(ls[buf ^ 1], voff, sbase);
            async_stage16(ld[buf ^ 1], voff, dbase);
            wait_async_le2();                // current buffer's 2 copies are done
        } else {
            wait_async_0();
        }

        int4 sv = stg_s[buf][tid];           // ds_load_b128
        int4 dv = stg_d[buf][tid];
        const int sj[EPT] = { sv.x, sv.y, sv.z, sv.w };
        const int di[EPT] = { dv.x, dv.y, dv.z, dv.w };

#pragma unroll
        for (int k = 0; k < EPT; ++k) {
            int j = sj[k], i = di[k];
            float4 pj = pk[j];               // {x,y,s,c} one b128 gather
            float4 pi = pk[i];
            // coef = ha[i] * sin(theta_j - theta_i)
            float coef = ha[i] * (pj.z * pi.w - pj.w * pi.z);
            unsafeAtomicAdd(&cxy[2 * (long long)i + 0], coef * (pj.x - pi.x));
            unsafeAtomicAdd(&cxy[2 * (long long)i + 1], coef * (pj.y - pi.y));
        }
        buf ^= 1;
        chunk = cnext;
    }
}

// ---------------------------------------------------------------
// Tail: edges not covered by full chunks (plain path, tiny).
// ---------------------------------------------------------------
__global__ void edge_tail_kernel(const int* __restrict__ esrc,
                                 const int* __restrict__ edst,
                                 const float4* __restrict__ pk,
                                 const float* __restrict__ ha,
                                 float* __restrict__ cxy,
                                 long long start, long long ne) {
    long long e = start + (long long)blockIdx.x * blockDim.x + threadIdx.x;
    long long stride = (long long)gridDim.x * blockDim.x;
    for (; e < ne; e += stride) {
        int j = esrc[e], i = edst[e];
        float4 pj = pk[j];
        float4 pi = pk[i];
        float coef = ha[i] * (pj.z * pi.w - pj.w * pi.z);
        unsafeAtomicAdd(&cxy[2 * (long long)i + 0], coef * (pj.x - pi.x));
        unsafeAtomicAdd(&cxy[2 * (long long)i + 1], coef * (pj.y - pi.y));
    }
}

// ---------------------------------------------------------------
// K3: finalize. Only y_new matters for the output angles.
// ---------------------------------------------------------------
__global__ void finalize_kernel(const float* __restrict__ x,
                                const float* __restrict__ y,
                                const float* __restrict__ w,
                                const float* __restrict__ amp,
                                const float* __restrict__ ph,
                                const float* __restrict__ b,
                                const float2* __restrict__ cxy,
                                float* __restrict__ out, int n) {
    int i = blockIdx.x * blockDim.x + threadIdx.x;
    if (i >= n) return;
    float xi = x[i], yi = y[i];
    float r2 = xi * xi + yi * yi + EPSF;
    float dy = (MUF - r2) * yi + w[i] * xi;
    float yn = yi + (dy + cxy[i].y) * DTF;
    float ang = amp[i] * yn + ph[i] + b[i];
    ang = fminf(fmaxf(ang, -BOUNDF), BOUNDF);
    out[i] = ang;
}

// ---------------------------------------------------------------
extern "C" void kernel_launch(void* const* d_in, const int* in_sizes, int n_in,
                              void* d_out, int out_size, void* d_ws, size_t ws_size,
                              hipStream_t stream) {
    const float* x    = (const float*)d_in[0];
    const float* y    = (const float*)d_in[1];
    const float* w    = (const float*)d_in[2];
    const float* amp  = (const float*)d_in[3];
    const float* ph   = (const float*)d_in[4];
    const float* ha   = (const float*)d_in[5];
    const float* b    = (const float*)d_in[6];
    const int*   esrc = (const int*)d_in[7];
    const int*   edst = (const int*)d_in[8];
    const int       n  = in_sizes[0];
    const long long ne = (long long)in_sizes[7];

    // workspace: packed node records (16B*n) then float2 accumulators (8B*n)
    float4* pk  = (float4*)d_ws;
    float2* cxy = (float2*)((char*)d_ws + (size_t)n * sizeof(float4));

    const int nb = (n + BLK - 1) / BLK;
    node_prep_kernel<<<nb, BLK, 0, stream>>>(x, y, pk, cxy, n);

    const long long nchunks = ne / CHUNK;
    if (nchunks > 0) {
        long long blocks = nchunks < 2048 ? nchunks : 2048;
        edge_pass_kernel<<<(int)blocks, BLK, 0, stream>>>(
            esrc, edst, pk, ha, (float*)cxy, nchunks);
    }
    const long long tail = nchunks * CHUNK;
    if (tail < ne) {
        edge_tail_kernel<<<64, BLK, 0, stream>>>(
            esrc, edst, pk, ha, (float*)cxy, tail, ne);
    }

    finalize_kernel<<<nb, BLK, 0, stream>>>(x, y, w, amp, ph, b, cxy,
                                            (float*)d_out, n);
}